// DeChunkLayerReference_38422777430601
// MI455X (gfx1250) — compile-verified
//
#include <hip/hip_runtime.h>
#include <hip/hip_bf16.h>
#include <math.h>

typedef __attribute__((ext_vector_type(2))) float v2f;
typedef __attribute__((ext_vector_type(8))) float v8f;

#define BB 4
#define LL 4096
#define MM 2048
#define DD 2048
#define NC 128          // number of 16-row chunks along M
#define EPSV 1e-4f

// ---- workspace layout (in floats) ----
static constexpr size_t Y_OFF     = 0;                                  // (B,M,D)
static constexpr size_t CARRY_OFF = Y_OFF     + (size_t)BB * MM * DD;   // (B,NC,D)
static constexpr size_t PS_OFF    = CARRY_OFF + (size_t)BB * NC * DD;   // (B,M)
static constexpr size_t DT_OFF    = PS_OFF    + (size_t)BB * MM;        // (B,M)
static constexpr size_t CV_OFF    = DT_OFF    + (size_t)BB * MM;        // (B,M)
static constexpr size_t AC_OFF    = CV_OFF    + (size_t)BB * MM;        // (B,NC)
static constexpr size_t PLUG_OFF  = AC_OFF    + (size_t)BB * NC;        // (B,L) int

// ---------------------------------------------------------------------------
// K1: per-batch block scan of boundary_mask; gather p at boundary positions,
//     compute dt = log(1/(1-p)), and plug_back = cumsum(mask)-1.
// ---------------------------------------------------------------------------
__global__ void prep_kernel(const float* __restrict__ bprob,
                            const unsigned char* __restrict__ bmask,
                            float* __restrict__ ws) {
  const int b   = blockIdx.x;
  const int tid = threadIdx.x;
  const int l0  = tid * 16;

  __shared__ int sc[256];

  int cnt = 0;
  unsigned char mloc[16];
  for (int e = 0; e < 16; ++e) {
    mloc[e] = bmask[(size_t)b * LL + l0 + e];
    cnt += mloc[e] ? 1 : 0;
  }
  sc[tid] = cnt;
  __syncthreads();
  for (int off = 1; off < 256; off <<= 1) {
    int v = (tid >= off) ? sc[tid - off] : 0;
    __syncthreads();
    sc[tid] += v;
    __syncthreads();
  }
  int running = sc[tid] - cnt;  // exclusive prefix

  float* p_s  = ws + PS_OFF + (size_t)b * MM;
  float* dt_s = ws + DT_OFF + (size_t)b * MM;
  int*   plug = (int*)(ws + PLUG_OFF) + (size_t)b * LL;

  for (int e = 0; e < 16; ++e) {
    const int l = l0 + e;
    if (mloc[e]) {
      if (running < MM) {
        float p = bprob[(size_t)b * LL + l];
        p = fminf(fmaxf(p, EPSV), 1.0f - EPSV);
        p_s[running]  = p;
        dt_s[running] = logf(1.0f / (1.0f - p));
      }
      ++running;
    }
    plug[l] = running - 1;
  }
}

// ---------------------------------------------------------------------------
// K2: chunked scan as WMMA matmul. One block per (b, chunk); 8 waves; each
//     wave handles 256 columns (16 tiles of 16) via 4x v_wmma_f32_16x16x4_f32.
//     dt roundtrip uses a precomputed per-row reciprocal (row-constant), so
//     the hot loop has no IEEE divide sequences.
// ---------------------------------------------------------------------------
__global__ void __launch_bounds__(256)
chunk_wmma_kernel(const float* __restrict__ hidden, float* __restrict__ ws) {
  const int blk = blockIdx.x;
  const int b   = blk / NC;
  const int c   = blk % NC;
  const int tid = threadIdx.x;

  __shared__ float sP[16], sA[16], sDt[16], sRdt[16];
  __shared__ float sM[16][17];  // lower-triangular chunk matrix (pad stride 17)

  if (tid < 16) {
    const float p  = ws[PS_OFF + (size_t)b * MM + c * 16 + tid];
    const float dt = ws[DT_OFF + (size_t)b * MM + c * 16 + tid];
    sP[tid]   = p;
    sA[tid]   = 1.0f - p;
    sDt[tid]  = dt;
    sRdt[tid] = 1.0f / dt;      // hoisted: row-constant reciprocal
  }
  __syncthreads();

  if (tid < 16) {
    const int i = tid;
    float run = 1.0f;
    for (int j = i; j >= 0; --j) {   // Mmat[i][j] = p[j] * prod_{k=j+1..i} a[k]
      sM[i][j] = sP[j] * run;
      run *= sA[j];
    }
    for (int j = i + 1; j < 16; ++j) sM[i][j] = 0.0f;
    // run == prod_{k=0..i} a[k] == cvec[i]
    ws[CV_OFF + (size_t)b * MM + c * 16 + i] = run;
    if (i == 15) ws[AC_OFF + (size_t)b * NC + c] = run;
  }
  __syncthreads();

  const int wave = tid >> 5;
  const int lane = tid & 31;
  const int half = lane >> 4;
  const int lrow = lane & 15;

  // A fragments (constant across column tiles). ISA layout: lanes 0-15 hold
  // K=0,1 (VGPR0,1); lanes 16-31 hold K=2,3 of each 4-wide K window.
  v2f afrag[4];
  int   rowsB[4][2];
  float dtsB[4][2];
  float rdtB[4][2];
  for (int kk = 0; kk < 4; ++kk) {
    const int k0 = kk * 4 + half * 2;
    afrag[kk][0] = sM[lrow][k0];
    afrag[kk][1] = sM[lrow][k0 + 1];
    rowsB[kk][0] = k0;
    rowsB[kk][1] = k0 + 1;
    dtsB[kk][0]  = sDt[k0];
    dtsB[kk][1]  = sDt[k0 + 1];
    rdtB[kk][0]  = sRdt[k0];
    rdtB[kk][1]  = sRdt[k0 + 1];
  }

  const float* hbase = hidden     + ((size_t)(b * MM + c * 16)) * DD;
  float*       ybase = ws + Y_OFF + ((size_t)(b * MM + c * 16)) * DD;
  const int colw = wave * 256;

  for (int tile = 0; tile < 16; ++tile) {
    const int col = colw + tile * 16 + lrow;

    v8f acc = {};  // carry handled later (gather adds cvec * carry)
    for (int kk = 0; kk < 4; ++kk) {
      float x0 = hbase[(size_t)rowsB[kk][0] * DD + col];
      float x1 = hbase[(size_t)rowsB[kk][1] * DD + col];
      // reference fp roundtrip hs = (hidden/dt)*dt, divide via reciprocal
      x0 = (x0 * rdtB[kk][0]) * dtsB[kk][0];
      x1 = (x1 * rdtB[kk][1]) * dtsB[kk][1];
      v2f bfrag;
      bfrag[0] = x0;
      bfrag[1] = x1;
      acc = __builtin_amdgcn_wmma_f32_16x16x4_f32(
          /*neg_a=*/false, afrag[kk], /*neg_b=*/false, bfrag,
          /*c_mod=*/(short)0, acc, /*reuse_a=*/false, /*reuse_b=*/false);
    }
    // C/D layout: VGPR r -> M = r + 8*half, N = lrow
#pragma unroll
    for (int r = 0; r < 8; ++r) {
      ybase[(size_t)(r + half * 8) * DD + col] = acc[r];
    }
  }
}

// ---------------------------------------------------------------------------
// K3: inter-chunk carry scan: carry[c] = A_{c-1}*carry[c-1] + y_local[c-1,15]
// ---------------------------------------------------------------------------
__global__ void carry_scan_kernel(float* __restrict__ ws) {
  const int b = blockIdx.y;
  const int d = blockIdx.x * 256 + threadIdx.x;

  const float* y  = ws + Y_OFF     + (size_t)b * MM * DD;
  float*    carry = ws + CARRY_OFF + (size_t)b * NC * DD;
  const float* Ac = ws + AC_OFF    + (size_t)b * NC;

  float h = 0.0f;
  for (int c = 0; c < NC; ++c) {
    carry[(size_t)c * DD + d] = h;
    h = Ac[c] * h + y[(size_t)(c * 16 + 15) * DD + d];
  }
}

// ---------------------------------------------------------------------------
// K4: plug back: out[b,l,:] = y[b,t,:] + cvec[b,t] * carry[b, t/16, :]
// ---------------------------------------------------------------------------
__global__ void gather_kernel(float* __restrict__ out,
                              const float* __restrict__ ws) {
  const int idx = blockIdx.x;
  const int b = idx / LL;
  const int l = idx % LL;

  const int* plug = (const int*)(ws + PLUG_OFF) + (size_t)b * LL;
  int t = plug[l];
  t = t < 0 ? 0 : (t >= MM ? MM - 1 : t);

  const float cv = ws[CV_OFF + (size_t)b * MM + t];
  const float4* yrow = (const float4*)(ws + Y_OFF     + ((size_t)(b * MM + t)) * DD);
  const float4* crow = (const float4*)(ws + CARRY_OFF + ((size_t)(b * NC + (t >> 4))) * DD);
  float4*       orow = (float4*)(out + ((size_t)(b * LL + l)) * DD);

  for (int i = threadIdx.x; i < DD / 4; i += 256) {
    const float4 yv = yrow[i];
    const float4 hv = crow[i];
    float4 o;
    o.x = yv.x + cv * hv.x;
    o.y = yv.y + cv * hv.y;
    o.z = yv.z + cv * hv.z;
    o.w = yv.w + cv * hv.w;
    orow[i] = o;
  }
}

extern "C" void kernel_launch(void* const* d_in, const int* in_sizes, int n_in,
                              void* d_out, int out_size, void* d_ws, size_t ws_size,
                              hipStream_t stream) {
  const float*         hidden = (const float*)d_in[0];          // (B, M, D) fp32
  const float*         bprob  = (const float*)d_in[1];          // (B, L)    fp32
  const unsigned char* bmask  = (const unsigned char*)d_in[2];  // (B, L)    bool
  // d_in[3] (mask) is unused by the reference.
  float* out = (float*)d_out;
  float* ws  = (float*)d_ws;

  prep_kernel<<<BB, 256, 0, stream>>>(bprob, bmask, ws);
  chunk_wmma_kernel<<<BB * NC, 256, 0, stream>>>(hidden, ws);
  {
    dim3 grid(DD / 256, BB);
    carry_scan_kernel<<<grid, 256, 0, stream>>>(ws);
  }
  gather_kernel<<<BB * LL, 256, 0, stream>>>(out, ws);
}